// Basemask_4836133175573
// MI455X (gfx1250) — compile-verified
//
#include <hip/hip_runtime.h>
#include <stdint.h>

// ---------------- problem constants (from setup_inputs) ----------------
#define NEG_INF_F (-1000000000.0f)
static constexpr int B_GRAPHS = 64;    // batch_size
static constexpr int NMAX     = 768;   // max_num_nodes
static constexpr int NHEADS   = 16;    // num_heads
static constexpr int FDIM     = 512;   // hidden dim
static constexpr int F4       = FDIM / 4;
static constexpr int ROWS_PER_GRAPH = NHEADS * NMAX;       // 12288 attn rows per graph
static constexpr int CHUNKS_PER_GRAPH = 16;
static constexpr int CH_ROWS  = ROWS_PER_GRAPH / CHUNKS_PER_GRAPH; // 768 rows per block

typedef __attribute__((ext_vector_type(4))) float v4f;
typedef __attribute__((ext_vector_type(4))) unsigned int u32x4;
typedef __attribute__((ext_vector_type(8))) int          i32x8;
typedef __attribute__((ext_vector_type(4))) int          i32x4;

// ---------------- kernel A: starts[b] = lower_bound(batch_ids, b) ----------------
__global__ void starts_kernel(const int* __restrict__ batch_ids, int N,
                              int* __restrict__ starts) {
    int t = threadIdx.x;
    if (t > B_GRAPHS) return;
    int lo = 0, hi = N;               // first index i with batch_ids[i] >= t
    while (lo < hi) {
        int mid = (lo + hi) >> 1;
        if (batch_ids[mid] < t) lo = mid + 1; else hi = mid;
    }
    starts[t] = lo;                   // starts[B] == N
}

// ---------------- kernel B: dense_x scatter (as gather) + Dmask ----------------
// one block per padded row (b,p); 128 threads x float4 = 512 floats
__global__ void __launch_bounds__(128)
dense_kernel(const v4f* __restrict__ x4, const int* __restrict__ starts,
             v4f* __restrict__ dense4, float* __restrict__ dmask) {
    int r = blockIdx.x;               // 0 .. B*NMAX-1
    int b = r / NMAX;
    int p = r - b * NMAX;
    int s0  = starts[b];
    int len = starts[b + 1] - s0;
    v4f v = (v4f)(0.0f);
    if (p < len) v = x4[(size_t)(s0 + p) * F4 + threadIdx.x];
    __builtin_nontemporal_store(v, &dense4[(size_t)r * F4 + threadIdx.x]);
    if (threadIdx.x == 0)
        __builtin_nontemporal_store((p < len) ? 1.0f : 0.0f, &dmask[r]);
}

// ---------------- kernel C: attn_mask broadcast (TDM iterate if available) ------
// grid = (CHUNKS_PER_GRAPH, B), block = 1 wave (32 threads).
// Row pattern for graph b: row[k] = (k < counts[b]) ? 0 : -1e9 ; identical for all
// (h, q) -> broadcast it 12288x via tensor_store_from_lds iterate mode.
__global__ void __launch_bounds__(32)
attn_kernel(const int* __restrict__ starts, float* __restrict__ attn) {
    __shared__ float row[NMAX];       // 3 KB row pattern, 16B aligned
    const int b     = blockIdx.y;
    const int chunk = blockIdx.x;
    const int len   = starts[b + 1] - starts[b];

    for (int k = threadIdx.x; k < NMAX; k += 32)
        row[k] = (k < len) ? 0.0f : NEG_INF_F;
    __syncthreads();
    asm volatile("s_wait_dscnt 0x0" ::: "memory");   // LDS row visible to TDM

    const size_t rowBase = (size_t)b * ROWS_PER_GRAPH + (size_t)chunk * CH_ROWS;
    float* gptr = attn + rowBase * NMAX;

#if defined(__has_builtin) && __has_builtin(__builtin_amdgcn_tensor_store_from_lds)
    const uint32_t lds_off = (uint32_t)(uintptr_t)&row[0]; // flat LDS addr[31:0] = LDS offset
    for (int it = 0; it < CH_ROWS; it += 256) {
        const int iters = (CH_ROWS - it) < 256 ? (CH_ROWS - it) : 256;
        const uint64_t ga = (uint64_t)(uintptr_t)(gptr + (size_t)it * NMAX);
        u32x4 g0; i32x8 g1; i32x4 g2, g3; i32x8 gx;
        // ---- D# group0: count=1 | lds_addr | global_addr(57b) | type=2 ----
        g0[0] = 1u;                                        // count=1 (valid), user mode
        g0[1] = lds_off;                                   // lds_addr
        g0[2] = (uint32_t)(ga & 0xFFFFFFFFu);              // global_addr[31:0]
        g0[3] = (uint32_t)((ga >> 32) & 0x1FFFFFFu) | (2u << 30); // addr[56:32] | type=2
        // ---- D# group1 ----
        g1[0] = (2 << 16) | (1 << 19);                     // data_size=4B, iterate_enable=1
        g1[1] = (NMAX & 0xFFFF) << 16;                     // tensor_dim0[15:0] @ [63:48]
        g1[2] = (1 << 16);                                 // dim0 hi=0 | tensor_dim1=1
        g1[3] = (NMAX & 0xFFFF) << 16;                     // dim1 hi=0 | tile_dim0=768
        g1[4] = 1;                                         // tile_dim1=1, tile_dim2 unused
        g1[5] = NMAX;                                      // tensor_dim0_stride lo32
        g1[6] = 0;                                         // stride0 hi | stride1 lo
        g1[7] = 0;                                         // stride1 hi
        // ---- D# group2 (iterate mode) ----
        g2[0] = 1;                                         // tensor_dim2 (unused)
        g2[1] = 0;                                         // lds_addr_increment = 0 (broadcast!)
        g2[2] = NMAX;                                      // global_addr_increment lo32 (elements)
        g2[3] = (iters - 1) << 16;                         // inc hi16=0 | iterate_count
        // ---- D# group3 ----
        g3[0] = 0; g3[1] = 0; g3[2] = 0; g3[3] = 0;
        // ---- extra 5th operand (clang-23 6-arg form): zero-filled, matches probe ----
        gx[0]=0; gx[1]=0; gx[2]=0; gx[3]=0; gx[4]=0; gx[5]=0; gx[6]=0; gx[7]=0;
        __builtin_amdgcn_tensor_store_from_lds(g0, g1, g2, g3, gx, 0);
    }
#if __has_builtin(__builtin_amdgcn_s_wait_tensorcnt)
    __builtin_amdgcn_s_wait_tensorcnt(0);                  // LDS must stay live for TDM
#else
    asm volatile("s_wait_tensorcnt 0x0" ::: "memory");
#endif
#else
    // Fallback: register-resident row, non-temporal b128 store stream.
    v4f vals[NMAX / (32 * 4)];                             // 6 float4 per lane
    #pragma unroll
    for (int j = 0; j < NMAX / 128; ++j) {
        int k = (j * 32 + threadIdx.x) * 4;
        v4f v; v.x = row[k]; v.y = row[k+1]; v.z = row[k+2]; v.w = row[k+3];
        vals[j] = v;
    }
    for (int r = 0; r < CH_ROWS; ++r) {
        v4f* dst = (v4f*)(gptr + (size_t)r * NMAX) + threadIdx.x;
        #pragma unroll
        for (int j = 0; j < NMAX / 128; ++j)
            __builtin_nontemporal_store(vals[j], dst + j * 32);
    }
#endif
}

// ---------------- host launcher ----------------
extern "C" void kernel_launch(void* const* d_in, const int* in_sizes, int n_in,
                              void* d_out, int out_size, void* d_ws, size_t ws_size,
                              hipStream_t stream) {
    const float* x         = (const float*)d_in[0];
    const int*   batch_ids = (const int*)d_in[1];
    const int N = in_sizes[1];                      // 32768

    int* starts = (int*)d_ws;                       // (B+1) ints of scratch

    float* out   = (float*)d_out;
    float* dense = out;                                                  // [B,NMAX,F]
    float* dmask = out + (size_t)B_GRAPHS * NMAX * FDIM;                 // [B,NMAX]
    float* attn  = dmask + (size_t)B_GRAPHS * NMAX;                      // [B,H,NMAX,NMAX]

    starts_kernel<<<1, 128, 0, stream>>>(batch_ids, N, starts);

    dense_kernel<<<B_GRAPHS * NMAX, 128, 0, stream>>>(
        (const v4f*)x, starts, (v4f*)dense, dmask);

    dim3 grid(CHUNKS_PER_GRAPH, B_GRAPHS);
    attn_kernel<<<grid, 32, 0, stream>>>(starts, attn);
}